// RigidBodyModule_29446295781709
// MI455X (gfx1250) — compile-verified
//
#include <hip/hip_runtime.h>
#include <math.h>

typedef float v2f __attribute__((ext_vector_type(2)));
typedef float v8f __attribute__((ext_vector_type(8)));

#define RB_EPS 0.1f

// D = A(16x4,f32) * B(4x16,f32) + C(16x16,f32), wave32 CDNA5 WMMA
static __device__ __forceinline__ v8f wmma4(v2f a, v2f b, v8f c) {
  return __builtin_amdgcn_wmma_f32_16x16x4_f32(false, a, false, b, (short)0, c, false, false);
}

// Branch-free tanh: 1 - 2/(e^{2x}+1); exact at +-inf, no EXEC divergence.
static __device__ __forceinline__ float fast_tanh(float x) {
  float e = __expf(2.f * x);                      // v_exp_f32 path
  return 1.f - 2.f * __builtin_amdgcn_rcpf(e + 1.f);
}

// 4 waves per block, 32 samples per wave, 128 samples per block.
__global__ void __launch_bounds__(128) rigid_body_kernel(
    const float* __restrict__ q, const float* __restrict__ v, const float* __restrict__ u,
    const float* __restrict__ W1, const float* __restrict__ b1, const float* __restrict__ W2,
    const float* __restrict__ Wp, const float* __restrict__ bp, const float* __restrict__ wv,
    float* __restrict__ out, int Btot)
{
  extern __shared__ float smem[];
  // ---- LDS carve (floats) ----
  float* W1s = smem;            // [16][64] rows 6..15 zero        1024
  float* Wps = W1s + 1024;      // [16][64]                        1024
  float* W2s = Wps + 1024;      // [64][48] cols 36..47 zero       3072
  float* W2p = W2s + 3072;      // [32][48][2] K-paired copy       3072
  float* b1s = W2p + 3072;      // [64]
  float* bps = b1s + 64;        // [64]
  float* wvs = bps + 64;        // [64]
  float* qb  = wvs + 64;        // [128][8] cols 6,7 zero          1024
  float* vb  = qb + 1024;       // [128][8]                        1024
  float* sH  = vb + 1024;       // 4 x [32][64]  H then CS         8192
  float* s2  = sH + 8192;       // 4 x [32][64]  SP -> VW -> SPV   8192
  float* sA  = s2 + 8192;       // 4 x [32][40]                    5120
  float* sD  = sA + 5120;       // 4 x [32][40]                    5120
  float* sKE = sD + 5120;       // 4 x [32][8]                     1024
  float* sG  = sKE + 1024;      // 4 x [32][8]                     1024
  // total = 39104 floats = 156416 bytes

  const int tid  = threadIdx.x;
  const int wave = tid >> 5;
  const int lane = tid & 31;
  const int half = lane >> 4;   // which 16-lane half
  const int ln   = lane & 15;
  const long long gbase = (long long)blockIdx.x * 128;

  // ---- stage weights + q/v tiles into LDS ----
  for (int i = tid; i < 1024; i += 128) {
    int r = i >> 6, c = i & 63;
    W1s[i] = (r < 6) ? W1[r*64+c] : 0.f;
    Wps[i] = (r < 6) ? Wp[r*64+c] : 0.f;
  }
  for (int i = tid; i < 3072; i += 128) {
    int k = i / 48, c = i % 48;
    float val = (c < 36) ? W2[k*36+c] : 0.f;
    W2s[i] = val;
    W2p[(k >> 1)*96 + c*2 + (k & 1)] = val;     // pair (k even, k odd) adjacent
  }
  if (tid < 64) { b1s[tid] = b1[tid]; bps[tid] = bp[tid]; wvs[tid] = wv[tid]; }
  for (int i = tid; i < 1024; i += 128) {
    int r = i >> 3, c = i & 7;
    long long s = gbase + r;
    float qv = 0.f, vv = 0.f;
    if (c < 6 && s < (long long)Btot) { qv = q[s*6+c]; vv = v[s*6+c]; }
    qb[i] = qv; vb[i] = vv;
  }
  __syncthreads();   // cross-wave producers above; everything below is per-wave

  float* wH  = sH  + wave*2048;
  float* w2  = s2  + wave*2048;
  float* wA  = sA  + wave*1280;
  float* wD  = sD  + wave*1280;
  float* wKE = sKE + wave*256;
  float* wG  = sG  + wave*256;
  float* wq  = qb  + wave*256;
  float* wvv = vb  + wave*256;

  // ===== Phase 1: H = tanh(Q@W1+b1), SP = (1-H^2)*(V@W1) =====
  for (int mt = 0; mt < 2; ++mt) {
    for (int nt = 0; nt < 4; ++nt) {
      float bb = b1s[nt*16 + ln];
      v8f hacc = {}, pacc = {};        // zero init -> inline-0 SRC2, bias folded below
      #pragma unroll
      for (int k0 = 0; k0 < 8; k0 += 4) {
        int krow = k0 + 2*half;
        v2f aq = *(const v2f*)&wq [(mt*16+ln)*8 + krow];
        v2f av = *(const v2f*)&wvv[(mt*16+ln)*8 + krow];
        v2f bw; bw.x = W1s[krow*64 + nt*16+ln]; bw.y = W1s[(krow+1)*64 + nt*16+ln];
        hacc = wmma4(aq, bw, hacc);
        pacc = wmma4(av, bw, pacc);
      }
      #pragma unroll
      for (int r = 0; r < 8; ++r) {
        int row = mt*16 + r + 8*half, col = nt*16 + ln;
        float h = fast_tanh(hacc[r] + bb);
        wH[row*64+col] = h;
        w2[row*64+col] = (1.f - h*h) * pacc[r];
      }
    }
  }
  __builtin_amdgcn_wave_barrier();

  // ===== Phase 2: A = H@W2 (32x36), D = SP@W2 =====
  for (int g = 0; g < 2; ++g) {
    const float* src = g ? w2 : wH;
    float* dst = g ? wD : wA;
    for (int mt = 0; mt < 2; ++mt) {
      for (int nt = 0; nt < 3; ++nt) {
        v8f acc = {};
        for (int k0 = 0; k0 < 64; k0 += 4) {
          int krow = k0 + 2*half;                 // always even
          v2f a = *(const v2f*)&src[(mt*16+ln)*64 + krow];
          v2f b = *(const v2f*)&W2p[(krow >> 1)*96 + (nt*16+ln)*2];
          acc = wmma4(a, b, acc);
        }
        int col = nt*16 + ln;
        if (col < 36) {
          #pragma unroll
          for (int r = 0; r < 8; ++r)
            dst[(mt*16 + r + 8*half)*40 + col] = acc[r];
        }
      }
    }
  }
  __builtin_amdgcn_wave_barrier();

  // ===== Phase 3 (per-lane, lane = sample): w = A^T v, VW = vec(v w^T) -> w2 =====
  {
    int j = lane;
    float vj[6], Ar[36], w_[6];
    #pragma unroll
    for (int i = 0; i < 6; ++i) vj[i] = wvv[j*8+i];
    #pragma unroll
    for (int c = 0; c < 36; ++c) Ar[c] = wA[j*40+c];
    #pragma unroll
    for (int c = 0; c < 6; ++c) {
      float s = 0.f;
      #pragma unroll
      for (int i = 0; i < 6; ++i) s += vj[i]*Ar[i*6+c];
      w_[c] = s;
    }
    #pragma unroll
    for (int i = 0; i < 6; ++i)
      #pragma unroll
      for (int c = 0; c < 6; ++c)
        w2[j*64 + i*6+c] = vj[i]*w_[c];
  }
  __builtin_amdgcn_wave_barrier();

  // ===== Phase 4: CS = (1-H^2) .* (VW @ W2^T), overwrite wH in place =====
  for (int nt = 0; nt < 4; ++nt) {
    for (int mt = 0; mt < 2; ++mt) {
      v8f acc = {};
      #pragma unroll
      for (int k0 = 0; k0 < 36; k0 += 4) {
        int krow = k0 + 2*half;
        v2f a = *(const v2f*)&w2[(mt*16+ln)*64 + krow];
        v2f b = *(const v2f*)&W2s[(nt*16+ln)*48 + krow];   // W2^T[k][n] = W2[n][k]
        acc = wmma4(a, b, acc);
      }
      #pragma unroll
      for (int r = 0; r < 8; ++r) {
        int row = mt*16 + r + 8*half, col = nt*16 + ln;
        float h = wH[row*64+col];
        wH[row*64+col] = (1.f - h*h) * acc[r];
      }
    }
  }
  __builtin_amdgcn_wave_barrier();

  // ===== Phase 5: Cv_KE = CS @ W1^T (32x6) =====
  for (int mt = 0; mt < 2; ++mt) {
    v8f acc = {};
    for (int k0 = 0; k0 < 64; k0 += 4) {
      int krow = k0 + 2*half;
      v2f a = *(const v2f*)&wH[(mt*16+ln)*64 + krow];
      v2f b = *(const v2f*)&W1s[ln*64 + krow];   // W1^T[k][n] = W1[n][k]; rows>=6 are zero
      acc = wmma4(a, b, acc);
    }
    #pragma unroll
    for (int r = 0; r < 8; ++r)
      if (ln < 8) wKE[(mt*16 + r + 8*half)*8 + ln] = acc[r];
  }
  __builtin_amdgcn_wave_barrier();

  // ===== Phase 6: SPV = (1-tanh(Q@Wp+bp)^2) .* wv  ->  G = SPV @ Wp^T =====
  for (int mt = 0; mt < 2; ++mt) {
    for (int nt = 0; nt < 4; ++nt) {
      float bb = bps[nt*16 + ln];
      v8f acc = {};
      #pragma unroll
      for (int k0 = 0; k0 < 8; k0 += 4) {
        int krow = k0 + 2*half;
        v2f aq = *(const v2f*)&wq[(mt*16+ln)*8 + krow];
        v2f bw; bw.x = Wps[krow*64 + nt*16+ln]; bw.y = Wps[(krow+1)*64 + nt*16+ln];
        acc = wmma4(aq, bw, acc);
      }
      #pragma unroll
      for (int r = 0; r < 8; ++r) {
        int row = mt*16 + r + 8*half, col = nt*16 + ln;
        float hp = fast_tanh(acc[r] + bb);
        w2[row*64+col] = (1.f - hp*hp) * wvs[col];
      }
    }
  }
  __builtin_amdgcn_wave_barrier();
  for (int mt = 0; mt < 2; ++mt) {
    v8f acc = {};
    for (int k0 = 0; k0 < 64; k0 += 4) {
      int krow = k0 + 2*half;
      v2f a = *(const v2f*)&w2[(mt*16+ln)*64 + krow];
      v2f b = *(const v2f*)&Wps[ln*64 + krow];   // Wp^T; rows>=6 zero
      acc = wmma4(a, b, acc);
    }
    #pragma unroll
    for (int r = 0; r < 8; ++r)
      if (ln < 8) wG[(mt*16 + r + 8*half)*8 + ln] = acc[r];
  }
  __builtin_amdgcn_wave_barrier();

  // ===== Phase 7 (per-lane): gMv, rhs, M = A A^T + eps I, Cholesky solve =====
  {
    int j = lane;
    long long s = gbase + wave*32 + j;
    if (s < (long long)Btot) {
      float vj[6], uj[6], Ar[36], Dr[36], w_[6], dv[6], rhs[6];
      #pragma unroll
      for (int i = 0; i < 6; ++i) { vj[i] = wvv[j*8+i]; uj[i] = u[s*6+i]; }
      #pragma unroll
      for (int c = 0; c < 36; ++c) { Ar[c] = wA[j*40+c]; Dr[c] = wD[j*40+c]; }
      #pragma unroll
      for (int c = 0; c < 6; ++c) {
        float sw = 0.f, sd = 0.f;
        #pragma unroll
        for (int i = 0; i < 6; ++i) { sw += vj[i]*Ar[i*6+c]; sd += vj[i]*Dr[i*6+c]; }
        w_[c] = sw; dv[c] = sd;
      }
      #pragma unroll
      for (int i = 0; i < 6; ++i) {
        float g = 0.f;
        #pragma unroll
        for (int c = 0; c < 6; ++c) g += Dr[i*6+c]*w_[c] + Ar[i*6+c]*dv[c];
        rhs[i] = uj[i] - g + wKE[j*8+i] - wG[j*8+i];
      }
      float L[21];                       // lower tri, idx(i,jj)=i*(i+1)/2+jj
      #pragma unroll
      for (int i = 0; i < 6; ++i)
        #pragma unroll
        for (int jj = 0; jj < 6; ++jj) {
          if (jj <= i) {
            float m = (i == jj) ? RB_EPS : 0.f;
            #pragma unroll
            for (int c = 0; c < 6; ++c) m += Ar[i*6+c]*Ar[jj*6+c];
            L[i*(i+1)/2 + jj] = m;
          }
        }
      #pragma unroll
      for (int c = 0; c < 6; ++c) {
        float d = L[c*(c+1)/2 + c];
        #pragma unroll
        for (int k = 0; k < c; ++k) { float lv = L[c*(c+1)/2+k]; d -= lv*lv; }
        d = sqrtf(d);
        L[c*(c+1)/2+c] = d;
        float inv = 1.f / d;
        #pragma unroll
        for (int i2 = c+1; i2 < 6; ++i2) {
          float sum = L[i2*(i2+1)/2 + c];
          #pragma unroll
          for (int k = 0; k < c; ++k) sum -= L[i2*(i2+1)/2+k]*L[c*(c+1)/2+k];
          L[i2*(i2+1)/2+c] = sum * inv;
        }
      }
      float y[6], x[6];
      #pragma unroll
      for (int i = 0; i < 6; ++i) {
        float sum = rhs[i];
        #pragma unroll
        for (int k = 0; k < i; ++k) sum -= L[i*(i+1)/2+k]*y[k];
        y[i] = sum / L[i*(i+1)/2+i];
      }
      #pragma unroll
      for (int i = 5; i >= 0; --i) {
        float sum = y[i];
        #pragma unroll
        for (int k = i+1; k < 6; ++k) sum -= L[k*(k+1)/2+i]*x[k];
        x[i] = sum / L[i*(i+1)/2+i];
      }
      long long off = (long long)Btot * 6;
      #pragma unroll
      for (int i = 0; i < 6; ++i) {
        out[s*6+i]       = vj[i];  // qdot = v
        out[off + s*6+i] = x[i];   // vdot
      }
    }
  }
}

extern "C" void kernel_launch(void* const* d_in, const int* in_sizes, int n_in,
                              void* d_out, int out_size, void* d_ws, size_t ws_size,
                              hipStream_t stream) {
  const float* q  = (const float*)d_in[0];
  const float* v  = (const float*)d_in[1];
  const float* u  = (const float*)d_in[2];
  const float* W1 = (const float*)d_in[3];
  const float* b1 = (const float*)d_in[4];
  const float* W2 = (const float*)d_in[5];
  const float* Wp = (const float*)d_in[6];
  const float* bp = (const float*)d_in[7];
  const float* wv = (const float*)d_in[8];
  // d_in[9] = t (unused)
  float* out = (float*)d_out;
  int Btot = in_sizes[0] / 6;
  int blocks = (Btot + 127) / 128;
  size_t shmem = 39104 * sizeof(float);   // ~153 KB of the 320 KB WGP LDS
  hipFuncSetAttribute((const void*)rigid_body_kernel,
                      hipFuncAttributeMaxDynamicSharedMemorySize, (int)shmem);
  hipLaunchKernelGGL(rigid_body_kernel, dim3(blocks), dim3(128), shmem, stream,
                     q, v, u, W1, b1, W2, Wp, bp, wv, out, Btot);
}